// Recommender_90469191123302
// MI455X (gfx1250) — compile-verified
//
#include <hip/hip_runtime.h>
#include <math.h>

typedef __attribute__((ext_vector_type(2))) float v2f;
typedef __attribute__((ext_vector_type(8))) float v8f;

#define DIM 64
#define N_ITEMS 50000
#define NEG_SLOPE 0.2f
#define NORM_EPS 1e-12f

// ---------- init: copy src into mutable "current" buffer and residual output ----------
__global__ void copy2_kernel(const float* __restrict__ src,
                             float* __restrict__ d1, float* __restrict__ d2, long n) {
    long i = (long)blockIdx.x * blockDim.x + threadIdx.x;
    if (i < n) { float v = src[i]; d1[i] = v; d2[i] = v; }
}

// ---------- edge scatter: agg[head] += ent[tail] * rel[etype]  (16 threads x float4 per edge) ----------
__global__ void edge_scatter_kernel(const float* __restrict__ ent,
                                    const float* __restrict__ rel,
                                    const int* __restrict__ head,
                                    const int* __restrict__ tail,
                                    const int* __restrict__ etype,
                                    float* __restrict__ agg, int n_edges) {
    long gid = (long)blockIdx.x * blockDim.x + threadIdx.x;
    long e = gid >> 4;
    if (e >= n_edges) return;
    int c = (int)(gid & 15) * 4;
    int t = tail[e], h = head[e], r = etype[e];
    const float4 ev = *(const float4*)(ent + (long)t * DIM + c);
    const float4 rv = *(const float4*)(rel + (long)r * DIM + c);
    float* dst = agg + (long)h * DIM + c;
    atomicAdd(dst + 0, ev.x * rv.x);
    atomicAdd(dst + 1, ev.y * rv.y);
    atomicAdd(dst + 2, ev.z * rv.z);
    atomicAdd(dst + 3, ev.w * rv.w);
}

// ---------- interaction scatter: iu_agg[mat_col] += user[mat_row] * rel[0] ----------
__global__ void inter_scatter_kernel(const float* __restrict__ usr,
                                     const float* __restrict__ rel,
                                     const int* __restrict__ mrow,
                                     const int* __restrict__ mcol,
                                     float* __restrict__ agg, int n_inter) {
    long gid = (long)blockIdx.x * blockDim.x + threadIdx.x;
    long i = gid >> 4;
    if (i >= n_inter) return;
    int c = (int)(gid & 15) * 4;
    int u = mrow[i], it = mcol[i];
    const float4 uv = *(const float4*)(usr + (long)u * DIM + c);
    const float4 rv = *(const float4*)(rel + c); // rel_weight[0]
    float* dst = agg + (long)it * DIM + c;
    atomicAdd(dst + 0, uv.x * rv.x);
    atomicAdd(dst + 1, uv.y * rv.y);
    atomicAdd(dst + 2, uv.z * rv.z);
    atomicAdd(dst + 3, uv.w * rv.w);
}

// ---------- WMMA GEMM + gated fusion ----------
// gi = leaky_relu(ea @ G1^T + iu @ G2^T); fus = gi*ea + (1-gi)*iu
// One wave per 16x16 output tile. K=64 in 16 steps of v_wmma_f32_16x16x4_f32.
// f32 A-layout (16x4): lanes 0-15 -> M=0..15, K = k0+{0,1}; lanes 16-31 -> K = k0+{2,3}.
// f32 B-layout (4x16): B[k][n] = W[n][k] (we need W^T), mirrored lane pattern -> same
// addressing as A with the gate-weight row index = output column.
__global__ void gemm_fusion_kernel(const float* __restrict__ ea,
                                   const float* __restrict__ iu,
                                   const float* __restrict__ g1,
                                   const float* __restrict__ g2,
                                   float* __restrict__ fus, int n_items) {
    int wave = blockIdx.x * (blockDim.x >> 5) + (threadIdx.x >> 5);
    int nRowTiles = (n_items + 15) >> 4;
    int rowTile = wave >> 2;
    int colTile = wave & 3;
    if (rowTile >= nRowTiles) return;   // whole-wave exit: EXEC stays all-ones for WMMA

    int lane = threadIdx.x & 31;
    int l16  = lane & 15;
    int hi   = lane >> 4;

    int m0 = rowTile << 4;
    int n0 = colTile << 4;

    v8f acc = {};
#pragma unroll
    for (int k0 = 0; k0 < DIM; k0 += 4) {
        int kk = k0 + 2 * hi;
        v2f a1 = *(const v2f*)(ea + (long)(m0 + l16) * DIM + kk);
        v2f b1 = *(const v2f*)(g1 + (long)(n0 + l16) * DIM + kk);
        acc = __builtin_amdgcn_wmma_f32_16x16x4_f32(false, a1, false, b1,
                                                    (short)0, acc, false, false);
        v2f a2 = *(const v2f*)(iu + (long)(m0 + l16) * DIM + kk);
        v2f b2 = *(const v2f*)(g2 + (long)(n0 + l16) * DIM + kk);
        acc = __builtin_amdgcn_wmma_f32_16x16x4_f32(false, a2, false, b2,
                                                    (short)0, acc, false, false);
    }

    // C/D layout: VGPR r, lanes 0-15 -> row m0+r, lanes 16-31 -> row m0+r+8; col = n0+l16
#pragma unroll
    for (int r = 0; r < 8; ++r) {
        long m = m0 + r + 8 * hi;
        long n = n0 + l16;
        float g = acc[r];
        g = (g > 0.0f) ? g : NEG_SLOPE * g;
        float e = ea[m * DIM + n];
        float u = iu[m * DIM + n];
        fus[m * DIM + n] = g * e + (1.0f - g) * u;
    }
}

// ---------- user scatter: user_agg[mat_row] += item_fusion[mat_col] ----------
__global__ void user_scatter_kernel(const float* __restrict__ fus,
                                    const int* __restrict__ mrow,
                                    const int* __restrict__ mcol,
                                    float* __restrict__ agg, int n_inter) {
    long gid = (long)blockIdx.x * blockDim.x + threadIdx.x;
    long i = gid >> 4;
    if (i >= n_inter) return;
    int c = (int)(gid & 15) * 4;
    int u = mrow[i], it = mcol[i];
    const float4 fv = *(const float4*)(fus + (long)it * DIM + c);
    float* dst = agg + (long)u * DIM + c;
    atomicAdd(dst + 0, fv.x);
    atomicAdd(dst + 1, fv.y);
    atomicAdd(dst + 2, fv.z);
    atomicAdd(dst + 3, fv.w);
}

// ---------- row-normalize + residual accumulate (one wave32 per row) ----------
__global__ void normalize_res_kernel(const float* __restrict__ srcA,   // rows < splitRow
                                     const float* __restrict__ srcB,   // rows >= splitRow
                                     int splitRow,
                                     float* __restrict__ cur,
                                     float* __restrict__ res, int n_rows) {
    int row  = blockIdx.x * (blockDim.x >> 5) + (threadIdx.x >> 5);
    int lane = threadIdx.x & 31;
    if (row >= n_rows) return;
    const float* src = (row < splitRow) ? srcA : srcB;
    long base = (long)row * DIM + lane * 2;
    float2 v = *(const float2*)(src + base);
    float s = v.x * v.x + v.y * v.y;
#pragma unroll
    for (int off = 16; off > 0; off >>= 1) s += __shfl_xor(s, off, 32);
    float scale = 1.0f / fmaxf(sqrtf(s), NORM_EPS);
    float2 o; o.x = v.x * scale; o.y = v.y * scale;
    *(float2*)(cur + base) = o;
    float2 rr = *(const float2*)(res + base);
    rr.x += o.x; rr.y += o.y;
    *(float2*)(res + base) = rr;
}

extern "C" void kernel_launch(void* const* d_in, const int* in_sizes, int n_in,
                              void* d_out, int out_size, void* d_ws, size_t ws_size,
                              hipStream_t stream) {
    const float* user_emb   = (const float*)d_in[0];
    const float* entity_emb = (const float*)d_in[1];
    const float* rel_weight = (const float*)d_in[2];
    const float* gate1_w    = (const float*)d_in[3];
    const float* gate2_w    = (const float*)d_in[4];
    const int*   head       = (const int*)d_in[5];
    const int*   tail       = (const int*)d_in[6];
    const int*   etype      = (const int*)d_in[7];
    const int*   mat_row    = (const int*)d_in[8];
    const int*   mat_col    = (const int*)d_in[9];

    const long n_users    = in_sizes[0] / DIM;
    const long n_entities = in_sizes[1] / DIM;
    const int  n_hops     = in_sizes[3] / (DIM * DIM);
    const long n_edges    = in_sizes[5];
    const long n_inter    = in_sizes[8];
    const long n_items    = N_ITEMS;

    float* ws = (float*)d_ws;
    long o = 0;
    float* ent_cur = ws + o; o += n_entities * DIM;
    float* ent_agg = ws + o; o += n_entities * DIM;
    float* fus     = ws + o; o += n_items * DIM;
    float* iu_agg  = ws + o; o += n_items * DIM;
    float* usr_cur = ws + o; o += n_users * DIM;
    float* usr_agg = ws + o; o += n_users * DIM;

    float* out_ent = (float*)d_out;
    float* out_usr = out_ent + n_entities * DIM;

    // residuals start at the input embeddings; current buffers are mutable copies
    {
        long n = n_entities * DIM;
        copy2_kernel<<<(int)((n + 255) / 256), 256, 0, stream>>>(entity_emb, ent_cur, out_ent, n);
        n = n_users * DIM;
        copy2_kernel<<<(int)((n + 255) / 256), 256, 0, stream>>>(user_emb, usr_cur, out_usr, n);
    }

    for (int hop = 0; hop < n_hops; ++hop) {
        hipMemsetAsync(ent_agg, 0, (size_t)n_entities * DIM * sizeof(float), stream);
        hipMemsetAsync(iu_agg,  0, (size_t)n_items   * DIM * sizeof(float), stream);
        hipMemsetAsync(usr_agg, 0, (size_t)n_users   * DIM * sizeof(float), stream);

        long tE = n_edges * 16;
        edge_scatter_kernel<<<(int)((tE + 255) / 256), 256, 0, stream>>>(
            ent_cur, rel_weight, head, tail, etype, ent_agg, (int)n_edges);

        long tI = n_inter * 16;
        inter_scatter_kernel<<<(int)((tI + 255) / 256), 256, 0, stream>>>(
            usr_cur, rel_weight, mat_row, mat_col, iu_agg, (int)n_inter);

        const float* g1 = gate1_w + (long)hop * DIM * DIM;
        const float* g2 = gate2_w + (long)hop * DIM * DIM;
        int nWaves  = (int)(((n_items + 15) / 16) * 4);
        int nBlocks = (nWaves + 7) / 8;              // 8 waves / 256-thread block
        gemm_fusion_kernel<<<nBlocks, 256, 0, stream>>>(ent_agg, iu_agg, g1, g2, fus, (int)n_items);

        user_scatter_kernel<<<(int)((tI + 255) / 256), 256, 0, stream>>>(
            fus, mat_row, mat_col, usr_agg, (int)n_inter);

        normalize_res_kernel<<<(int)((n_entities + 7) / 8), 256, 0, stream>>>(
            fus, ent_agg, (int)n_items, ent_cur, out_ent, (int)n_entities);
        normalize_res_kernel<<<(int)((n_users + 7) / 8), 256, 0, stream>>>(
            usr_agg, usr_agg, 0, usr_cur, out_usr, (int)n_users);
    }
}